// RelGraphConvLayer_49392123904721
// MI455X (gfx1250) — compile-verified
//
#include <hip/hip_runtime.h>

typedef __attribute__((ext_vector_type(2))) float v2f;
typedef __attribute__((ext_vector_type(8))) float v8f;

#define DIM 256           // IN_DIM == OUT_DIM == 256
#define NREL 8
#define KCHUNK 32
#define ROWS_PER_BLK 128  // 8 waves x 16 rows
#define LDS_STRIDE 36     // 32 + 4 pad: (j*36)%64 distinct for j=0..15 -> no bank conflicts

// ---------------------------------------------------------------------------
// out[n, o] = bias[o]   (out must be re-initialized every call; it is poisoned)
// ---------------------------------------------------------------------------
__global__ void relconv_init(float* __restrict__ out, const float* __restrict__ bias,
                             int total4) {
  const float4* b4 = (const float4*)bias;          // 64 float4 = 256 floats
  float4* o4 = (float4*)out;
  for (int i = blockIdx.x * blockDim.x + threadIdx.x; i < total4;
       i += gridDim.x * blockDim.x) {
    o4[i] = b4[i & 63];
  }
}

// ---------------------------------------------------------------------------
// H[n, o] = sum_i X[n, i] * Wr[o, i]      (Wr = weight[r], row-major [out][in])
// 8 waves per block; each wave owns 16 rows x 256 cols (16 WMMA accumulators).
// A fragment (16x4 f32):  lanes 0-15 hold {K0,K1}, lanes 16-31 hold {K2,K3}.
// B fragment (4x16 f32):  lane j holds column o=j: {K0,K1} / {K2,K3}.
// C/D (16x16 f32): VGPR v -> M = v (+8 for lanes>=16), N = lane&15.
// ---------------------------------------------------------------------------
__global__ __launch_bounds__(256, 1)
void relconv_gemm(const float* __restrict__ X, const float* __restrict__ Wr,
                  float* __restrict__ H, int N) {
  __shared__ float Xs[ROWS_PER_BLK][LDS_STRIDE];
  __shared__ float Ws[DIM][LDS_STRIDE];

  const int tid  = threadIdx.x;
  const int wave = tid >> 5;
  const int lane = tid & 31;
  const int row0 = blockIdx.x * ROWS_PER_BLK;
  const int wrow = row0 + wave * 16;           // N % 16 == 0 -> wave all-valid or all-skip
  const bool active = (wrow < N);

  v8f acc[16];
#pragma unroll
  for (int i = 0; i < 16; ++i) acc[i] = (v8f){0.f, 0.f, 0.f, 0.f, 0.f, 0.f, 0.f, 0.f};

  const int mlane = lane & 15;
  const int ksel  = (lane >= 16) ? 2 : 0;
  const int arow  = wave * 16 + mlane;

  for (int kc = 0; kc < DIM; kc += KCHUNK) {
    __syncthreads();
    // ---- cooperative stage: X tile 128x32 (1024 float4, 4 per thread) ----
    {
      int f4 = tid * 4;
#pragma unroll
      for (int u = 0; u < 4; ++u, ++f4) {
        const int r  = f4 >> 3;                // tile row
        const int c4 = f4 & 7;                 // float4 within row
        int gr = row0 + r;
        gr = (gr < N) ? gr : (N - 1);          // clamp: garbage only feeds skipped waves
        float4 v = *(const float4*)(X + (size_t)gr * DIM + kc + c4 * 4);
        *(float4*)&Xs[r][c4 * 4] = v;
      }
    }
    // ---- cooperative stage: W chunk 256x32 (2048 float4, 8 per thread) ----
    {
      int f4 = tid * 8;
#pragma unroll
      for (int u = 0; u < 8; ++u, ++f4) {
        const int o  = f4 >> 3;
        const int c4 = f4 & 7;
        float4 v = *(const float4*)(Wr + (size_t)o * DIM + kc + c4 * 4);
        *(float4*)&Ws[o][c4 * 4] = v;
      }
    }
    __syncthreads();

    if (active) {
#pragma unroll
      for (int kk = 0; kk < KCHUNK; kk += 4) {
        const v2f a = *(const v2f*)&Xs[arow][kk + ksel];   // loop-invariant over ct
#pragma unroll
        for (int ct = 0; ct < 16; ++ct) {
          const v2f b = *(const v2f*)&Ws[ct * 16 + mlane][kk + ksel];
          acc[ct] = __builtin_amdgcn_wmma_f32_16x16x4_f32(
              false, a, false, b, (short)0, acc[ct], false, false);
        }
      }
    }
  }

  if (active) {
    const int radd = (lane >= 16) ? 8 : 0;
#pragma unroll
    for (int ct = 0; ct < 16; ++ct) {
#pragma unroll
      for (int v = 0; v < 8; ++v) {
        H[(size_t)(wrow + radd + v) * DIM + ct * 16 + mlane] = acc[ct][v];
      }
    }
  }
}

// ---------------------------------------------------------------------------
// For every edge e with edge_types[e] == rel:
//   out[dst[e], :] += H[src[e], :]
// One wave per edge: 2x b128 gather (L2-resident H), 8 f32 atomics per lane
// (L2-resident out). Both 51 MB buffers fit the 192 MB L2 together.
// ---------------------------------------------------------------------------
__global__ __launch_bounds__(256)
void relconv_scatter(const float* __restrict__ H, const int* __restrict__ ei,
                     const int* __restrict__ et, float* __restrict__ out,
                     int E, int rel) {
  const int lane = threadIdx.x & 31;
  const int wid  = (blockIdx.x * blockDim.x + threadIdx.x) >> 5;
  const int nw   = (gridDim.x * blockDim.x) >> 5;

  for (int e = wid; e < E; e += nw) {
    if (et[e] != rel) continue;                 // wave-uniform branch
    const int s = ei[e];
    const int d = ei[E + e];
    const float4* hp = (const float4*)(H + (size_t)s * DIM);
    const float4 a = hp[lane];                  // floats [lane*4 .. +3]
    const float4 b = hp[lane + 32];             // floats [128 + lane*4 .. +3]
    // hint the next matching edge's source row toward L2 (global_prefetch_b8)
    int en = e + nw;
    if (en < E && et[en] == rel) {
      __builtin_prefetch(H + (size_t)ei[en] * DIM + lane * 4, 0, 1);
    }
    float* op = out + (size_t)d * DIM + lane * 4;
    atomicAdd(op + 0, a.x);
    atomicAdd(op + 1, a.y);
    atomicAdd(op + 2, a.z);
    atomicAdd(op + 3, a.w);
    atomicAdd(op + 128 + 0, b.x);
    atomicAdd(op + 128 + 1, b.y);
    atomicAdd(op + 128 + 2, b.z);
    atomicAdd(op + 128 + 3, b.w);
  }
}

// ---------------------------------------------------------------------------
// Fallback (only if d_ws cannot hold one [N,256] f32 buffer): direct per-edge
// GEMV with LDS-staged source row. 2x the FLOPs but no workspace.
// ---------------------------------------------------------------------------
__global__ __launch_bounds__(256, 2)
void relconv_edge_gemv(const float* __restrict__ X, const float* __restrict__ W,
                       const int* __restrict__ ei, const int* __restrict__ et,
                       float* __restrict__ out, int E) {
  __shared__ float xs[DIM];
  const int o = threadIdx.x;
  for (int e = blockIdx.x; e < E; e += gridDim.x) {
    const int t = et[e];
    const int s = ei[e];
    const int d = ei[E + e];
    __syncthreads();
    xs[o] = X[(size_t)s * DIM + o];
    __syncthreads();
    const float4* wr = (const float4*)(W + ((size_t)t * DIM + o) * DIM);
    const float4* xv = (const float4*)xs;
    float sum = 0.f;
#pragma unroll 8
    for (int i = 0; i < DIM / 4; ++i) {
      const float4 w4 = wr[i];
      const float4 x4 = xv[i];
      sum = fmaf(w4.x, x4.x, sum);
      sum = fmaf(w4.y, x4.y, sum);
      sum = fmaf(w4.z, x4.z, sum);
      sum = fmaf(w4.w, x4.w, sum);
    }
    atomicAdd(out + (size_t)d * DIM + o, sum);
  }
}

// ---------------------------------------------------------------------------
extern "C" void kernel_launch(void* const* d_in, const int* in_sizes, int n_in,
                              void* d_out, int out_size, void* d_ws, size_t ws_size,
                              hipStream_t stream) {
  const float* X    = (const float*)d_in[0];   // [N, 256] f32
  const float* W    = (const float*)d_in[1];   // [8, 256, 256] f32
  const float* bias = (const float*)d_in[2];   // [256] f32
  const int*   ei   = (const int*)d_in[3];     // [2, E] (harness delivers int32)
  const int*   et   = (const int*)d_in[4];     // [E]
  float*       out  = (float*)d_out;           // [N, 256] f32

  const int N = in_sizes[0] / DIM;
  const int E = in_sizes[4];

  // out = bias (broadcast); out is poisoned before timing, so always re-init.
  relconv_init<<<1024, 256, 0, stream>>>(out, bias, N * (DIM / 4));

  const size_t need = (size_t)N * DIM * sizeof(float);
  if (ws_size >= need) {
    float* H = (float*)d_ws;
    const int gblocks = (N + ROWS_PER_BLK - 1) / ROWS_PER_BLK;
    for (int r = 0; r < NREL; ++r) {
      relconv_gemm<<<gblocks, 256, 0, stream>>>(X, W + (size_t)r * DIM * DIM, H, N);
      relconv_scatter<<<4096, 256, 0, stream>>>(H, ei, et, out, E, r);
    }
  } else {
    relconv_edge_gemv<<<8192, 256, 0, stream>>>(X, W, ei, et, out, E);
  }
}